// OneUnkMaterialSparseStack_NK_79061757984795
// MI455X (gfx1250) — compile-verified
//
#include <hip/hip_runtime.h>
#include <math.h>

#ifndef __has_builtin
#define __has_builtin(x) 0
#endif

#define PW   200   // table points
#define WN   1024  // wavelengths
#define AN   512   // angles
#define NL   13    // layers
#define BLK  256

// ---------------------------------------------------------------- complex f32
struct cx { float x, y; };
__device__ __forceinline__ cx  mkc(float a, float b)     { cx r; r.x = a; r.y = b; return r; }
__device__ __forceinline__ cx  cadd(cx a, cx b)          { return mkc(a.x + b.x, a.y + b.y); }
__device__ __forceinline__ cx  csub(cx a, cx b)          { return mkc(a.x - b.x, a.y - b.y); }
__device__ __forceinline__ cx  cmul(cx a, cx b)          { return mkc(a.x*b.x - a.y*b.y, a.x*b.y + a.y*b.x); }
__device__ __forceinline__ cx  cscale(cx a, float s)     { return mkc(a.x*s, a.y*s); }
__device__ __forceinline__ float cnorm2(cx a)            { return a.x*a.x + a.y*a.y; }
__device__ __forceinline__ cx  cdiv(cx a, cx b) {
  float inv = 1.0f / (b.x*b.x + b.y*b.y);
  return mkc((a.x*b.x + a.y*b.y)*inv, (a.y*b.x - a.x*b.y)*inv);
}
// principal complex sqrt (matches numpy/jnp branch)
__device__ __forceinline__ cx csqrtc(cx z) {
  float m  = sqrtf(z.x*z.x + z.y*z.y);
  float re = sqrtf(fmaxf(0.5f*(m + z.x), 0.0f));
  float im = sqrtf(fmaxf(0.5f*(m - z.x), 0.0f));
  return mkc(re, copysignf(im, z.y));
}

// ------------------------------------------------ CDNA5 async global->LDS copy
// Builtin signature (from clang diagnostic): param0 = v2i in AS1 (global),
// param1 = v2i in AS3 (LDS), then imm offset, imm cpol.
#if __has_builtin(__builtin_amdgcn_global_load_async_to_lds_b64)
#define HAVE_ASYNC_LDS 1
typedef int v2i __attribute__((ext_vector_type(2)));
typedef v2i __attribute__((address_space(1)))* g_v2i_ptr;
typedef v2i __attribute__((address_space(3)))* l_v2i_ptr;
__device__ __forceinline__ void lds_async_copy8(void* lds_dst, const void* gsrc) {
  // AS1 (global) pointer: bit-identical to the flat/global generic address.
  // AS3 (LDS) pointer: low 32 bits of the generic shared-aperture address are
  // the LDS byte offset.
  __builtin_amdgcn_global_load_async_to_lds_b64(
      (g_v2i_ptr)(uintptr_t)gsrc,
      (l_v2i_ptr)(unsigned int)(uintptr_t)lds_dst,
      0, 0);
}
__device__ __forceinline__ void lds_async_wait0() {
#if __has_builtin(__builtin_amdgcn_s_wait_asynccnt)
  __builtin_amdgcn_s_wait_asynccnt(0);
#else
  asm volatile("s_wait_asynccnt 0" ::: "memory");
#endif
}
#else
#define HAVE_ASYNC_LDS 0
#endif

// ---------------------------------------------------------------- kernel 1
// Per-(layer, wavelength) linear interpolation of (n, k) from material tables.
__global__ __launch_bounds__(256) void k_build_ntab(
    const float* __restrict__ fixed_data,   // [3][3][PW]
    const float* __restrict__ dyn_wl,       // [PW]
    const float* __restrict__ dyn_n,        // [PW]
    const float* __restrict__ dyn_k,        // [PW]
    const int*   __restrict__ matdist,      // [NL]
    const float* __restrict__ wavelengths,  // [WN]
    float2*      __restrict__ ntab)         // [NL][WN] (re, im)
{
  int idx = blockIdx.x * blockDim.x + threadIdx.x;
  if (idx >= NL * WN) return;
  int l = idx >> 10;          // idx / WN
  int w = idx & (WN - 1);
  float x = wavelengths[w];
  int m = matdist[l];
  const float *xs, *yn, *yk;
  if (m < 3) { const float* b = fixed_data + m * 3 * PW; xs = b; yn = b + PW; yk = b + 2 * PW; }
  else       { xs = dyn_wl; yn = dyn_n; yk = dyn_k; }

  float nr, nk;
  if (x <= xs[0])            { nr = yn[0];      nk = yk[0];      }
  else if (x >= xs[PW - 1])  { nr = yn[PW - 1]; nk = yk[PW - 1]; }
  else {
    int lo = 0, hi = PW - 1;
    #pragma unroll 1
    while (hi - lo > 1) { int mid = (lo + hi) >> 1; if (xs[mid] <= x) lo = mid; else hi = mid; }
    float t = (x - xs[lo]) / (xs[hi] - xs[lo]);
    nr = yn[lo] + t * (yn[hi] - yn[lo]);
    nk = yk[lo] + t * (yk[hi] - yk[lo]);
  }
  ntab[idx] = make_float2(nr, nk);
}

// ---------------------------------------------------------------- kernel 2
// One thread per (angle, wavelength). lambda contiguous per lane -> coalesced.
// Block = 256 lambdas at a fixed angle; layer-index slice staged in LDS via
// the CDNA5 async global->LDS path.
__global__ __launch_bounds__(BLK) void k_tmm(
    const float2* __restrict__ ntab,        // [NL][WN]
    const float*  __restrict__ wavelengths, // [WN]
    const float*  __restrict__ aoi,         // [AN]
    const float*  __restrict__ th_above,    // [5]
    const float*  __restrict__ unk_thick,   // [1] (nm)
    const float*  __restrict__ th_below,    // [5]
    float*        __restrict__ out)         // [3][AN][WN]
{
  __shared__ float2 nlds[NL * BLK];         // 26 KB of the 320 KB WGP LDS
  const int t   = threadIdx.x;
  const int bid = blockIdx.x;
  const int a   = bid >> 2;                 // 4 wavelength tiles per angle
  const int w   = ((bid & 3) << 8) + t;

#if HAVE_ASYNC_LDS
  #pragma unroll
  for (int l = 0; l < NL; ++l)
    lds_async_copy8(&nlds[l * BLK + t], &ntab[l * WN + w]);
  lds_async_wait0();
#else
  #pragma unroll
  for (int l = 0; l < NL; ++l)
    nlds[l * BLK + t] = ntab[l * WN + w];
#endif
  __syncthreads();

  // interior-layer thicknesses (microns); fully-unrolled chain keeps these in regs
  float dth[11];
  #pragma unroll
  for (int i = 0; i < 5; ++i) dth[i] = th_above[i];
  dth[5] = unk_thick[0] * 1e-3f;
  #pragma unroll
  for (int i = 0; i < 5; ++i) dth[6 + i] = th_below[i];

  const float lam  = wavelengths[w];
  const float kk0  = 6.2831853071795864f / lam;   // 2*pi/lambda
  const float sin0 = sinf(aoi[a]);

  // layer 0
  cx n0; { float2 v = nlds[t]; n0 = mkc(v.x, v.y); }
  cx s0  = cscale(n0, sin0);                       // n0 * sin(theta0)
  cx sj0 = cdiv(s0, n0);
  cx c0  = csqrtc(csub(mkc(1.0f, 0.0f), cmul(sj0, sj0)));
  const float fs_den = n0.x * c0.x - n0.y * c0.y;  // Re(n0 * c0)
  const float fp_den = n0.x * c0.x + n0.y * c0.y;  // Re(n0 * conj(c0))

  cx np_ = n0, cp_ = c0;                           // previous layer n, cos
  cx Ms00 = mkc(0,0), Ms01 = mkc(0,0), Ms10 = mkc(0,0), Ms11 = mkc(0,0), Pts = mkc(1,0);
  cx Mp00 = mkc(0,0), Mp01 = mkc(0,0), Mp10 = mkc(0,0), Mp11 = mkc(0,0), Ptp = mkc(1,0);

  #pragma unroll
  for (int l = 1; l < NL; ++l) {
    cx nl; { float2 v = nlds[l * BLK + t]; nl = mkc(v.x, v.y); }
    cx sj = cdiv(s0, nl);
    cx cl = csqrtc(csub(mkc(1.0f, 0.0f), cmul(sj, sj)));

    cx nici = cmul(np_, cp_);
    cx njcj = cmul(nl,  cl);
    cx nicj = cmul(np_, cl);
    cx njci = cmul(nl,  cp_);

    cx den_s = cadd(nici, njcj);
    cx rs    = cdiv(csub(nici, njcj), den_s);
    cx ts    = cdiv(cscale(nici, 2.0f), den_s);
    cx den_p = cadd(njci, nicj);
    cx rp    = cdiv(csub(njci, nicj), den_p);
    cx tp    = cdiv(cscale(nici, 2.0f), den_p);

    if (l == 1) {
      // D0 with 1/t0 factored into Pt
      Ms00 = mkc(1,0); Ms01 = rs; Ms10 = rs; Ms11 = mkc(1,0); Pts = ts;
      Mp00 = mkc(1,0); Mp01 = rp; Mp10 = rp; Mp11 = mkc(1,0); Ptp = tp;
    } else {
      // layer (l-1) phase: delta = (2pi/lambda) * n_{l-1} * cos_{l-1} * d_{l-1}
      cx delta = cscale(nici, kk0 * dth[l - 2]);
      float ey = expf(delta.y);
      float sn, cs2; sincosf(delta.x, &sn, &cs2);
      float iey = 1.0f / ey;
      cx em = mkc( ey * cs2, -ey * sn);   // exp(-i*delta)
      cx ep = mkc(iey * cs2, iey * sn);   // exp(+i*delta)
      // s-pol: M <- M * [[em, em*r],[ep*r, ep]]  (1/t factored into Pt)
      {
        cx emr = cmul(em, rs), epr = cmul(ep, rs);
        cx t00 = cadd(cmul(Ms00, em ), cmul(Ms01, epr));
        cx t01 = cadd(cmul(Ms00, emr), cmul(Ms01, ep ));
        cx t10 = cadd(cmul(Ms10, em ), cmul(Ms11, epr));
        cx t11 = cadd(cmul(Ms10, emr), cmul(Ms11, ep ));
        Ms00 = t00; Ms01 = t01; Ms10 = t10; Ms11 = t11;
        Pts  = cmul(Pts, ts);
      }
      // p-pol
      {
        cx emr = cmul(em, rp), epr = cmul(ep, rp);
        cx t00 = cadd(cmul(Mp00, em ), cmul(Mp01, epr));
        cx t01 = cadd(cmul(Mp00, emr), cmul(Mp01, ep ));
        cx t10 = cadd(cmul(Mp10, em ), cmul(Mp11, epr));
        cx t11 = cadd(cmul(Mp10, emr), cmul(Mp11, ep ));
        Mp00 = t00; Mp01 = t01; Mp10 = t10; Mp11 = t11;
        Ptp  = cmul(Ptp, tp);
      }
    }
    np_ = nl; cp_ = cl;
  }

  // rr = m10'/m00', tt = Pt/m00'  (scalar 1/t prefactors cancel / fold into Pt)
  const float fs_num = np_.x * cp_.x - np_.y * cp_.y;  // Re(n_last * c_last)
  const float fp_num = np_.x * cp_.x + np_.y * cp_.y;  // Re(n_last * conj(c_last))
  const float im00s = 1.0f / cnorm2(Ms00);
  const float im00p = 1.0f / cnorm2(Mp00);
  const float Rs = cnorm2(Ms10) * im00s;
  const float Ts = cnorm2(Pts)  * im00s * (fs_num / fs_den);
  const float Rp = cnorm2(Mp10) * im00p;
  const float Tp = cnorm2(Ptp)  * im00p * (fp_num / fp_den);

  const float R = 0.5f * (Rs + Rp);
  const float T = 0.5f * (Ts + Tp);
  const int o = a * WN + w;
  out[o]               = R;
  out[AN * WN + o]     = T;
  out[2 * AN * WN + o] = 1.0f - R - T;
}

// ---------------------------------------------------------------- launch
extern "C" void kernel_launch(void* const* d_in, const int* in_sizes, int n_in,
                              void* d_out, int out_size, void* d_ws, size_t ws_size,
                              hipStream_t stream) {
  const float* refractive_index = (const float*)d_in[0]; // [200]
  const float* extinction       = (const float*)d_in[1]; // [200]
  const float* unk_thick        = (const float*)d_in[2]; // [1]
  const float* fixed_data       = (const float*)d_in[3]; // [3][3][200]
  const float* dyn_wl           = (const float*)d_in[4]; // [200]
  const int*   matdist          = (const int*)  d_in[5]; // [13]
  const float* th_above         = (const float*)d_in[6]; // [5]
  const float* th_below         = (const float*)d_in[7]; // [5]
  const float* wavelengths      = (const float*)d_in[8]; // [1024]
  const float* aoi              = (const float*)d_in[9]; // [512]
  float*  out  = (float*)d_out;
  float2* ntab = (float2*)d_ws;                          // 13*1024*8 B = 104 KB

  k_build_ntab<<<(NL * WN + 255) / 256, 256, 0, stream>>>(
      fixed_data, dyn_wl, refractive_index, extinction, matdist, wavelengths, ntab);

  k_tmm<<<AN * (WN / BLK), BLK, 0, stream>>>(
      ntab, wavelengths, aoi, th_above, unk_thick, th_below, out);
}